// GaussianRasterizer_20435454394442
// MI455X (gfx1250) — compile-verified
//
#include <hip/hip_runtime.h>

#define N_G   512
#define H_IMG 256
#define W_IMG 256
#define HW    (H_IMG * W_IMG)
#define KC    32              // gaussians per blend chunk
#define FXC   256.0f          // W / (2*tanfovx)
#define FYC   256.0f          // H / (2*tanfovy)
#define TANF  0.5f

typedef __attribute__((ext_vector_type(2))) float v2f;
typedef __attribute__((ext_vector_type(8))) float v8f;

// ---------------------------------------------------------------------------
// Kernel 1: per-gaussian preprocessing + bitonic depth sort (one block, 512 thr)
// ws layout (floats): U[11][512] unsorted {depth,px,py,ca,cb,cc,op,r,g,b,invd}
//                     S[10][512] sorted   {px,py,ca,cb,cc,op,r,g,b,invd}
// ---------------------------------------------------------------------------
__global__ __launch_bounds__(N_G) void gs_preprocess_sort(
    const float* __restrict__ means3D, const float* __restrict__ opac,
    const float* __restrict__ colors,  const float* __restrict__ scales,
    const float* __restrict__ rots,    const float* __restrict__ vm,
    const float* __restrict__ pm,      float* __restrict__ radii_out,
    float* __restrict__ ws)
{
    __shared__ float skey[N_G];
    __shared__ int   sidx[N_G];
    const int g = threadIdx.x;
    float* U = ws;
    float* S = ws + 11 * N_G;

    {
        const float x = means3D[g*3+0], y = means3D[g*3+1], z = means3D[g*3+2];
        const float t0 = vm[0]*x + vm[1]*y + vm[2]*z  + vm[3];
        const float t1 = vm[4]*x + vm[5]*y + vm[6]*z  + vm[7];
        const float t2 = vm[8]*x + vm[9]*y + vm[10]*z + vm[11];
        const float depth = t2;
        const float c0 = pm[0]*x  + pm[1]*y  + pm[2]*z  + pm[3];
        const float c1 = pm[4]*x  + pm[5]*y  + pm[6]*z  + pm[7];
        const float c3 = pm[12]*x + pm[13]*y + pm[14]*z + pm[15];
        const float pw = 1.0f / (c3 + 1e-7f);
        const float px = ((c0*pw + 1.0f) * (float)W_IMG - 1.0f) * 0.5f;
        const float py = ((c1*pw + 1.0f) * (float)H_IMG - 1.0f) * 0.5f;

        float qw = rots[g*4+0], qx = rots[g*4+1], qy = rots[g*4+2], qz = rots[g*4+3];
        const float qn = rsqrtf(qw*qw + qx*qx + qy*qy + qz*qz);
        qw *= qn; qx *= qn; qy *= qn; qz *= qn;
        float R[3][3];
        R[0][0] = 1.f - 2.f*(qy*qy + qz*qz); R[0][1] = 2.f*(qx*qy - qw*qz); R[0][2] = 2.f*(qx*qz + qw*qy);
        R[1][0] = 2.f*(qx*qy + qw*qz); R[1][1] = 1.f - 2.f*(qx*qx + qz*qz); R[1][2] = 2.f*(qy*qz - qw*qx);
        R[2][0] = 2.f*(qx*qz - qw*qy); R[2][1] = 2.f*(qy*qz + qw*qx); R[2][2] = 1.f - 2.f*(qx*qx + qy*qy);
        const float s0 = scales[g*3+0], s1 = scales[g*3+1], s2 = scales[g*3+2];
        const float s0s = s0*s0, s1s = s1*s1, s2s = s2*s2;
        float C3[3][3];
        #pragma unroll
        for (int i = 0; i < 3; ++i)
            #pragma unroll
            for (int k = 0; k < 3; ++k)
                C3[i][k] = s0s*R[i][0]*R[k][0] + s1s*R[i][1]*R[k][1] + s2s*R[i][2]*R[k][2];

        const float tz = depth;
        const float limx = 1.3f * TANF, limy = 1.3f * TANF;
        const float txv = fminf(fmaxf(t0 / tz, -limx), limx) * tz;
        const float tyv = fminf(fmaxf(t1 / tz, -limy), limy) * tz;
        const float J00 = FXC / tz, J02 = -FXC * txv / (tz * tz);
        const float J11 = FYC / tz, J12 = -FYC * tyv / (tz * tz);
        float T0[3], T1[3];
        #pragma unroll
        for (int c = 0; c < 3; ++c) {
            T0[c] = J00 * vm[0*4 + c] + J02 * vm[2*4 + c];
            T1[c] = J11 * vm[1*4 + c] + J12 * vm[2*4 + c];
        }
        float A0[3], A1[3];
        #pragma unroll
        for (int k = 0; k < 3; ++k) {
            A0[k] = T0[0]*C3[0][k] + T0[1]*C3[1][k] + T0[2]*C3[2][k];
            A1[k] = T1[0]*C3[0][k] + T1[1]*C3[1][k] + T1[2]*C3[2][k];
        }
        const float a = A0[0]*T0[0] + A0[1]*T0[1] + A0[2]*T0[2] + 0.3f;
        const float b = A0[0]*T1[0] + A0[1]*T1[1] + A0[2]*T1[2];
        const float c = A1[0]*T1[0] + A1[1]*T1[1] + A1[2]*T1[2] + 0.3f;
        const float det = a*c - b*b;
        const float invdet = 1.0f / ((det != 0.0f) ? det : 1.0f);
        const float mid = 0.5f * (a + c);
        const float lam = mid + sqrtf(fmaxf(mid*mid - det, 0.1f));
        const int   radii = (int)ceilf(3.0f * sqrtf(lam));
        const bool  valid = (depth > 0.2f) && (det > 0.0f);

        radii_out[g] = (float)radii;   // f32 output buffer (JAX promotion)

        U[0*N_G + g]  = depth;
        U[1*N_G + g]  = px;
        U[2*N_G + g]  = py;
        U[3*N_G + g]  =  c * invdet;   // ca
        U[4*N_G + g]  = -b * invdet;   // cb
        U[5*N_G + g]  =  a * invdet;   // cc
        U[6*N_G + g]  = valid ? opac[g] : 0.0f;
        U[7*N_G + g]  = colors[g*3+0];
        U[8*N_G + g]  = colors[g*3+1];
        U[9*N_G + g]  = colors[g*3+2];
        U[10*N_G + g] = 1.0f / fmaxf(depth, 1e-6f);
        skey[g] = depth;
        sidx[g] = g;
    }
    __syncthreads();

    // bitonic sort ascending on depth, payload = index
    for (int k = 2; k <= N_G; k <<= 1) {
        for (int j = k >> 1; j > 0; j >>= 1) {
            const int i = threadIdx.x;
            const int l = i ^ j;
            if (l > i) {
                const bool  up = ((i & k) == 0);
                const float ki = skey[i], kl = skey[l];
                const bool  sw = up ? (ki > kl) : (ki < kl);
                if (sw) {
                    skey[i] = kl; skey[l] = ki;
                    const int tmp = sidx[i]; sidx[i] = sidx[l]; sidx[l] = tmp;
                }
            }
            __syncthreads();
        }
    }
    const int src = sidx[threadIdx.x];
    #pragma unroll
    for (int a2 = 0; a2 < 10; ++a2)
        S[a2 * N_G + threadIdx.x] = U[(a2 + 1) * N_G + src];
}

// ---------------------------------------------------------------------------
// Kernel 2: tiled alpha blending. 256 blocks x 256 threads; thread = pixel.
// Sequential transmittance in registers; weighted color/invd accumulation via
// V_WMMA_F32_16X16X4_F32 (A = 16px x 4 wgt, B = 4 x {r,g,b,invd,0...}).
// ---------------------------------------------------------------------------
__global__ __launch_bounds__(256) void gs_blend(
    const float* __restrict__ ws, const float* __restrict__ bgv,
    const float* __restrict__ mask, float* __restrict__ out)
{
    const float* S = ws + 11 * N_G;
    __shared__ float sp[10 * N_G];      // sorted params       (20 KB)
    __shared__ float wgt[256 * 33];     // per-chunk weights   (33.8 KB, padded)
    __shared__ float res[256 * 4];      // D-matrix scatter    (4 KB)

    const int t = threadIdx.x;
    for (int i = t; i < 10 * N_G; i += 256) sp[i] = S[i];
    const float* spx = sp + 0*N_G; const float* spy = sp + 1*N_G;
    const float* sca = sp + 2*N_G; const float* scb = sp + 3*N_G;
    const float* scc = sp + 4*N_G; const float* sop = sp + 5*N_G;
    const float* sr  = sp + 6*N_G; const float* sgc = sp + 7*N_G;
    const float* sbc = sp + 8*N_G; const float* sid = sp + 9*N_G;

    const int tileX = blockIdx.x & 15, tileY = blockIdx.x >> 4;
    const int gx = tileX * 16 + (t & 15);
    const int gy = tileY * 16 + (t >> 4);
    const float fx = (float)gx, fy = (float)gy;

    const int lane = t & 31;
    const int wave = t >> 5;
    const int n    = lane & 15;            // D/B column for this lane
    const int ksel = (lane < 16) ? 0 : 2;  // f32 A/B: lanes 16-31 hold K+2
    const int pix0 = wave * 32 + n;        // A tile0 row -> pixel
    const int pix1 = pix0 + 16;            // A tile1 row -> pixel

    float T = 1.0f;
    v8f c0 = {};
    v8f c1 = {};
    __syncthreads();

    for (int chunk = 0; chunk < N_G / KC; ++chunk) {
        const int base = chunk * KC;
        float* wrow = &wgt[t * 33];
        // sequential front-to-back blending for this pixel, chunk slab -> LDS
        #pragma unroll 4
        for (int k = 0; k < KC; ++k) {
            const int gI = base + k;
            const float dx = fx - spx[gI], dy = fy - spy[gI];
            const float power = -0.5f * (sca[gI]*dx*dx + scc[gI]*dy*dy) - scb[gI]*dx*dy;
            float al = fminf(0.99f, sop[gI] * __expf(fminf(power, 0.0f)));
            al = (power <= 0.0f && al >= (1.0f / 255.0f)) ? al : 0.0f;
            wrow[k] = al * T;
            T *= (1.0f - al);
        }
        __syncthreads();
        // WMMA accumulation: D(16px x 16ch) += A(16x4 wgt) * B(4x16 channels)
        #pragma unroll
        for (int kk = 0; kk < KC / 4; ++kk) {
            const int kb = kk * 4 + ksel;
            v2f a0, a1, bf;
            a0.x = wgt[pix0 * 33 + kb];     a0.y = wgt[pix0 * 33 + kb + 1];
            a1.x = wgt[pix1 * 33 + kb];     a1.y = wgt[pix1 * 33 + kb + 1];
            const int g0 = base + kb;
            bf.x = (n == 0) ? sr[g0]   : (n == 1) ? sgc[g0]   : (n == 2) ? sbc[g0]   : (n == 3) ? sid[g0]   : 0.0f;
            bf.y = (n == 0) ? sr[g0+1] : (n == 1) ? sgc[g0+1] : (n == 2) ? sbc[g0+1] : (n == 3) ? sid[g0+1] : 0.0f;
            c0 = __builtin_amdgcn_wmma_f32_16x16x4_f32(false, a0, false, bf, (short)0, c0, false, false);
            c1 = __builtin_amdgcn_wmma_f32_16x16x4_f32(false, a1, false, bf, (short)0, c1, false, false);
        }
        __syncthreads();
    }

    // scatter D accumulators (VGPR v -> M=v / M=v+8) to per-pixel channels
    #pragma unroll
    for (int v = 0; v < 8; ++v) {
        const int m = (lane < 16) ? v : (v + 8);
        if (n < 4) {
            res[(wave * 32 + m) * 4 + n]      = c0[v];
            res[(wave * 32 + 16 + m) * 4 + n] = c1[v];
        }
    }
    __syncthreads();

    const float mk = mask[gy * W_IMG + gx];
    const float aR = res[t*4+0], aG = res[t*4+1], aB = res[t*4+2], aD = res[t*4+3];
    out[0*HW + gy*W_IMG + gx] = (aR + T * bgv[0]) * mk;
    out[1*HW + gy*W_IMG + gx] = (aG + T * bgv[1]) * mk;
    out[2*HW + gy*W_IMG + gx] = (aB + T * bgv[2]) * mk;
    out[3*HW + N_G + gy*W_IMG + gx] = aD * mk;
}

// ---------------------------------------------------------------------------
extern "C" void kernel_launch(void* const* d_in, const int* in_sizes, int n_in,
                              void* d_out, int out_size, void* d_ws, size_t ws_size,
                              hipStream_t stream) {
    const float* means3D = (const float*)d_in[0];
    // d_in[1] means2D unused (zeros in reference)
    const float* opac    = (const float*)d_in[2];
    const float* colors  = (const float*)d_in[3];
    const float* scales  = (const float*)d_in[4];
    const float* rots    = (const float*)d_in[5];
    const float* vm      = (const float*)d_in[6];
    const float* pm      = (const float*)d_in[7];
    const float* bg      = (const float*)d_in[8];
    const float* mask    = (const float*)d_in[9];
    float* out = (float*)d_out;
    float* ws  = (float*)d_ws;

    gs_preprocess_sort<<<1, N_G, 0, stream>>>(means3D, opac, colors, scales, rots,
                                              vm, pm, out + 3 * HW, ws);
    gs_blend<<<256, 256, 0, stream>>>(ws, bg, mask, out);
}